// OCWCCE_2637109919779
// MI455X (gfx1250) — compile-verified
//
#include <hip/hip_runtime.h>
#include <hip/hip_bf16.h>

// Problem geometry (hard-coded to the reference):
//   B=32768 rows, T=256 timepoints, C=3 classes.
// Kernel1: 2048 blocks x 256 threads, 16 rows (4096 positions) per block,
// staged as 4 chunks of 1024 positions through double-buffered LDS using
// CDNA5 async global->LDS copies (ASYNCcnt).

#define NBLK1   2048
#define ROWS_PB 16
#define POS_PB  4096   // ROWS_PB * 256
#define CHUNKP  1024   // positions per chunk
#define NCHUNK  4      // POS_PB / CHUNKP

template <int N>
__device__ __forceinline__ void wait_async() {
#if __has_builtin(__builtin_amdgcn_s_wait_asynccnt)
    __builtin_amdgcn_s_wait_asynccnt(N);
#else
    asm volatile("s_wait_asynccnt %0" :: "n"(N) : "memory");
#endif
}

// 16-byte async copy global -> LDS (gfx1250 GLOBAL_LOAD_ASYNC_TO_LDS_B128, GV mode).
__device__ __forceinline__ void async_copy16(unsigned lds_off, const void* gaddr) {
    asm volatile("global_load_async_to_lds_b128 %0, %1, off"
                 :: "v"(lds_off), "v"(gaddr)
                 : "memory");
}

// One (b,t) position: cce = logsumexp(x) - x[target], masked by validity.
__device__ __forceinline__ void proc_pos(float tg, float x0, float x1, float x2,
                                         float& acc, float& cnt) {
    const bool v = (tg >= 0.0f);              // padding is -inf
    const float y0 = v ? x0 : 0.0f;
    const float y1 = v ? x1 : 0.0f;
    const float y2 = v ? x2 : 0.0f;
    const float m  = fmaxf(y0, fmaxf(y1, y2));
    const float es = __expf(y0 - m) + __expf(y1 - m) + __expf(y2 - m);
    const float lse = m + __logf(es);
    const float ts = v ? tg : 0.0f;
    const int   ti = (int)ts;
    const float chosen = (ti == 0) ? y0 : ((ti == 1) ? y1 : y2);
    acc += v ? (lse - chosen) : 0.0f;
    cnt += v ? 1.0f : 0.0f;
}

__global__ __launch_bounds__(256)
void ocwcce_partial_kernel(const float* __restrict__ logits,
                           const float* __restrict__ targets,
                           float* __restrict__ partS,   // [NBLK1][256]
                           float* __restrict__ partC) { // [NBLK1][256]
    __shared__ float lbuf[2][3072];   // 12 KB logits per chunk, double buffered
    __shared__ float tbuf[2][1024];   //  4 KB targets per chunk
    __shared__ float red_s[256];
    __shared__ float red_c[256];

    const int tid = threadIdx.x;
    const size_t blockPos = (size_t)blockIdx.x * POS_PB;

    auto issue = [&](int c, int b) {
        const size_t posBase = blockPos + (size_t)c * CHUNKP;
        const char* gl = (const char*)logits  + posBase * 12; // 3 floats/pos
        const char* gt = (const char*)targets + posBase * 4;  // 1 float/pos
        const unsigned ldsL = (unsigned)(unsigned long long)&lbuf[b][0];
        const unsigned ldsT = (unsigned)(unsigned long long)&tbuf[b][0];
        async_copy16(ldsL +          tid * 16u, gl +        tid * 16);
        async_copy16(ldsL + 4096u +  tid * 16u, gl + 4096 + tid * 16);
        async_copy16(ldsL + 8192u +  tid * 16u, gl + 8192 + tid * 16);
        async_copy16(ldsT +          tid * 16u, gt +        tid * 16);
    };

    float acc0 = 0.f, acc1 = 0.f, acc2 = 0.f, acc3 = 0.f;
    float c0 = 0.f, c1 = 0.f, c2 = 0.f, c3 = 0.f;

    issue(0, 0);
    issue(1, 1);

#pragma unroll
    for (int k = 0; k < NCHUNK; ++k) {
        // Loads complete in order: waiting <=4 with up to 8 outstanding means
        // the current chunk's 4 ops are done.
        if (k < NCHUNK - 1) wait_async<4>(); else wait_async<0>();
        __syncthreads();   // all waves' chunk-k data now resident in LDS

        const int b = k & 1;
        const float4 tg = *(const float4*)&tbuf[b][tid * 4];
        const float* lp = &lbuf[b][tid * 12];
        const float4 p = *(const float4*)(lp + 0);
        const float4 q = *(const float4*)(lp + 4);
        const float4 r = *(const float4*)(lp + 8);

        proc_pos(tg.x, p.x, p.y, p.z, acc0, c0);
        proc_pos(tg.y, p.w, q.x, q.y, acc1, c1);
        proc_pos(tg.z, q.z, q.w, r.x, acc2, c2);
        proc_pos(tg.w, r.y, r.z, r.w, acc3, c3);

        __syncthreads();   // everyone finished reading buffer b
        if (k + 2 < NCHUNK) issue(k + 2, b);   // safe to overwrite
    }

    // Deterministic per-block combine: threads tid, tid+64, tid+128, tid+192
    // share the same 4-t strip t0..t0+3; merge them in 4 fixed phases.
    red_s[tid] = 0.0f;
    red_c[tid] = 0.0f;
    __syncthreads();
    const int t0 = (tid & 63) * 4;
#pragma unroll
    for (int ph = 0; ph < 4; ++ph) {
        if ((tid >> 6) == ph) {
            red_s[t0 + 0] += acc0; red_s[t0 + 1] += acc1;
            red_s[t0 + 2] += acc2; red_s[t0 + 3] += acc3;
            red_c[t0 + 0] += c0;   red_c[t0 + 1] += c1;
            red_c[t0 + 2] += c2;   red_c[t0 + 3] += c3;
        }
        __syncthreads();
    }

    partS[(size_t)blockIdx.x * 256 + tid] = red_s[tid];
    partC[(size_t)blockIdx.x * 256 + tid] = red_c[tid];
}

// One block per t: tree-reduce 2048 partial sums/counts -> ratio[t].
__global__ __launch_bounds__(256)
void ocwcce_reduce_t_kernel(const float* __restrict__ partS,
                            const float* __restrict__ partC,
                            float* __restrict__ ratio) {
    __shared__ float ss[256];
    __shared__ float cs[256];
    const int t = blockIdx.x;
    const int i = threadIdx.x;
    float s = 0.f, c = 0.f;
#pragma unroll
    for (int m = 0; m < NBLK1 / 256; ++m) {
        const size_t j = (size_t)(i + m * 256) * 256 + t;
        s += partS[j];
        c += partC[j];
    }
    ss[i] = s; cs[i] = c;
    __syncthreads();
    for (int o = 128; o > 0; o >>= 1) {
        if (i < o) { ss[i] += ss[i + o]; cs[i] += cs[i + o]; }
        __syncthreads();
    }
    if (i == 0) ratio[t] = ss[0] / cs[0];
}

// Final mean over the 256 per-t ratios.
__global__ __launch_bounds__(256)
void ocwcce_final_kernel(const float* __restrict__ ratio,
                         float* __restrict__ out) {
    __shared__ float ss[256];
    const int i = threadIdx.x;
    ss[i] = ratio[i];
    __syncthreads();
    for (int o = 128; o > 0; o >>= 1) {
        if (i < o) ss[i] += ss[i + o];
        __syncthreads();
    }
    if (i == 0) out[0] = ss[0] * (1.0f / 256.0f);
}

extern "C" void kernel_launch(void* const* d_in, const int* in_sizes, int n_in,
                              void* d_out, int out_size, void* d_ws, size_t ws_size,
                              hipStream_t stream) {
    const float* logits  = (const float*)d_in[0];   // [32768,256,3] f32
    const float* targets = (const float*)d_in[1];   // [32768,256]   f32 (-inf pad)
    float* out = (float*)d_out;
    float* ws  = (float*)d_ws;

    float* partS = ws;                                   // 2048*256 floats
    float* partC = ws + (size_t)NBLK1 * 256;             // 2048*256 floats
    float* ratio = ws + (size_t)2 * NBLK1 * 256;         // 256 floats
    (void)in_sizes; (void)n_in; (void)out_size; (void)ws_size;

    ocwcce_partial_kernel<<<NBLK1, 256, 0, stream>>>(logits, targets, partS, partC);
    ocwcce_reduce_t_kernel<<<256, 256, 0, stream>>>(partS, partC, ratio);
    ocwcce_final_kernel<<<1, 256, 0, stream>>>(ratio, out);
}